// Conformer_2920577761292
// MI455X (gfx1250) — compile-verified
//
#include <hip/hip_runtime.h>

typedef __attribute__((ext_vector_type(2))) float v2f;
typedef __attribute__((ext_vector_type(4))) float f4;
typedef __attribute__((ext_vector_type(8))) float v8f;

#define B_DIM 2
#define T_DIM 512
#define U_DIM 128
#define D_DIM 512
#define V_DIM 1024   // vocab
#define W_LD  (2 * D_DIM)

// -----------------------------------------------------------------------------
// Phase 1: P[m][n] = sum_k A[m][k] * W[n][woff + k]
//   A: M x 512 row-major, W: 1024 x 1024 row-major, P: M x 1024 row-major
// Block = 256 threads = 8 waves. Wave w of block bx computes rows
// [16*bx, 16*bx+16) x cols [128*w, 128*w+128)  (8 WMMA 16x16 tiles).
// K is swept in steps of 4 with V_WMMA_F32_16X16X4_F32 (exact fp32 math).
// -----------------------------------------------------------------------------
__global__ __launch_bounds__(256) void joint_gemm_wmma(
    const float* __restrict__ A,
    const float* __restrict__ W,
    float* __restrict__ P,
    int woff)
{
  const int lane  = threadIdx.x & 31;
  const int wave  = threadIdx.x >> 5;
  const int l15   = lane & 15;   // M index for A-frag, N index for B-frag
  const int kh    = lane >> 4;   // selects K pair {2kh, 2kh+1} within the 4-step
  const int tileM = blockIdx.x;
  const int nbase = wave * 128;

  // A fragment: lane l15 holds row (16*tileM + l15), K elements 2kh,2kh+1
  const float* ap = A + (size_t)(tileM * 16 + l15) * D_DIM + 2 * kh;
  // B fragment: lane l15 holds column n = nbase + j*16 + l15; B[k][n] = W[n][woff+k]
  const float* wp = W + (size_t)(nbase + l15) * W_LD + woff + 2 * kh;

  v8f acc[8] = {};

  for (int k0 = 0; k0 < D_DIM; k0 += 4) {
    v2f a = *(const v2f*)ap;
    ap += 4;
#pragma unroll
    for (int j = 0; j < 8; ++j) {
      // j*16 rows of W  ->  j*65536 bytes: folds into the load's 24-bit offset
      v2f b = *(const v2f*)(wp + (size_t)j * 16 * W_LD);
      acc[j] = __builtin_amdgcn_wmma_f32_16x16x4_f32(
          /*neg_a=*/false, a, /*neg_b=*/false, b,
          /*c_mod=*/(short)0, acc[j], /*reuse_a=*/false, /*reuse_b=*/false);
    }
    wp += 4;
  }

  // C/D layout: VGPR r -> row r (lanes 0-15) / row r+8 (lanes 16-31), col = l15
  float* prow = P + (size_t)(tileM * 16 + kh * 8) * V_DIM + nbase + l15;
#pragma unroll
  for (int j = 0; j < 8; ++j) {
#pragma unroll
    for (int r = 0; r < 8; ++r) {
      prow[(size_t)r * V_DIM + j * 16] = acc[j][r];
    }
  }
}

// -----------------------------------------------------------------------------
// Phase 2 (bandwidth-dominant): out[b,t,u,v] = E[b*T+t, v] + Dp[b*U+u, v]
// One block per (b,t) row: each thread keeps its enc float4 in registers and
// loops over all U=128 decoder rows, writing 128-bit non-temporal stores
// (537 MB output >> 192 MB L2, so don't let it thrash the cache).
// -----------------------------------------------------------------------------
__global__ __launch_bounds__(256) void joint_bcast_add(
    const f4* __restrict__ E,   // (B*T) x 256 float4
    const f4* __restrict__ Dp,  // (B*U) x 256 float4
    f4* __restrict__ O)         // (B*T*U) x 256 float4
{
  const int bt = blockIdx.x;        // 0 .. B*T-1
  const int b  = bt >> 9;           // T = 512
  const int v4 = threadIdx.x;       // 0 .. 255 (V/4 = 256)

  const f4 e = E[(size_t)bt * (V_DIM / 4) + v4];
  const f4* drow = Dp + (size_t)b * U_DIM * (V_DIM / 4) + v4;
  f4* orow = O + (size_t)bt * U_DIM * (V_DIM / 4) + v4;

#pragma unroll 4
  for (int u = 0; u < U_DIM; ++u) {
    f4 d = drow[(size_t)u * (V_DIM / 4)];
    f4 r = e + d;
    __builtin_nontemporal_store(r, &orow[(size_t)u * (V_DIM / 4)]);
  }
}

// -----------------------------------------------------------------------------
extern "C" void kernel_launch(void* const* d_in, const int* in_sizes, int n_in,
                              void* d_out, int out_size, void* d_ws, size_t ws_size,
                              hipStream_t stream) {
  const float* enc = (const float*)d_in[0];  // (B, T, D) fp32
  const float* dec = (const float*)d_in[1];  // (B, U, D) fp32
  const float* W   = (const float*)d_in[2];  // (V, 2D) fp32
  float* out = (float*)d_out;                // (B, T, U, V) fp32

  float* Pe = (float*)d_ws;                     // enc_proj: 1024 x 1024 fp32 (4 MB)
  float* Pd = Pe + (size_t)(B_DIM * T_DIM) * V_DIM;  // dec_proj: 256 x 1024 fp32 (1 MB)

  // enc_proj: M = B*T = 1024 -> 64 blocks of 8 waves
  joint_gemm_wmma<<<(B_DIM * T_DIM) / 16, 256, 0, stream>>>(enc, W, Pe, /*woff=*/0);
  // dec_proj: M = B*U = 256 -> 16 blocks
  joint_gemm_wmma<<<(B_DIM * U_DIM) / 16, 256, 0, stream>>>(dec, W, Pd, /*woff=*/D_DIM);
  // broadcast add, one block per (b,t)
  joint_bcast_add<<<B_DIM * T_DIM, 256, 0, stream>>>(
      (const f4*)Pe, (const f4*)Pd, (f4*)out);
}